// GravityField_82497731821985
// MI455X (gfx1250) — compile-verified
//
#include <hip/hip_runtime.h>
#include <math.h>

typedef __attribute__((ext_vector_type(2))) float v2f;
typedef __attribute__((ext_vector_type(4))) float v4f;
typedef __attribute__((ext_vector_type(8))) float v8f;

#define ND 64            // D
#define NRFF 64          // N_RFF
#define WPB 8            // waves per block (256 threads)
#define RFF_C 0.17677669529663687f  // sqrt(2/64)

__device__ __forceinline__ float softplus_f(float x) {
  // stable softplus matching jax.nn.softplus
  return fmaxf(x, 0.0f) + log1pf(expf(-fabsf(x)));
}

__global__ void init_kernel(float* __restrict__ p) {
  p[threadIdx.x] = 0.0f;
}

// ---------------- pass 1: mass + phi, reduce phi_sum[b][64] ----------------
__global__ __launch_bounds__(256) void pass1_kernel(
    const float* __restrict__ coords,
    const float* __restrict__ w1,  const float* __restrict__ b1,
    const float* __restrict__ w2,  const float* __restrict__ b2,
    const float* __restrict__ rffW, const float* __restrict__ rffb,
    float* __restrict__ phi_sum)
{
  __shared__ float lds_sum[NRFF];
  __shared__ float lds_mass[WPB][16];

  const int tid = threadIdx.x;
  if (tid < NRFF) lds_sum[tid] = 0.0f;
  __syncthreads();

  const int lane = tid & 31;
  const int wave = tid >> 5;
  const int tile = blockIdx.x * WPB + wave;       // 16 tokens per tile
  const long tok0 = (long)tile * 16;
  const int m    = lane & 15;
  const int koff = (lane >= 16) ? 2 : 0;

  // coords tile in WMMA A-layout: lane holds row m, K = 4k + koff + {0,1}
  const float* crow = coords + (tok0 + (long)m) * ND + koff;
  v2f A[16];
#pragma unroll
  for (int k = 0; k < 16; ++k)
    A[k] = *(const v2f*)(crow + 4 * k);

  // ---- h = relu(coords @ w1 + b1); rowpart = partial h @ w2 ----
  float rowpart[8];
#pragma unroll
  for (int i = 0; i < 8; ++i) rowpart[i] = 0.0f;

#pragma unroll
  for (int n = 0; n < 4; ++n) {
    const int col = n * 16 + m;
    const float bias = b1[col];
    v8f acc;
#pragma unroll
    for (int i = 0; i < 8; ++i) acc[i] = bias;
#pragma unroll
    for (int k = 0; k < 16; ++k) {
      const int kr = 4 * k + koff;
      v2f Bv;
      Bv.x = w1[kr * ND + col];
      Bv.y = w1[(kr + 1) * ND + col];
      acc = __builtin_amdgcn_wmma_f32_16x16x4_f32(
          false, A[k], false, Bv, (short)0, acc, false, false);
    }
    const float w2c = w2[col];
#pragma unroll
    for (int i = 0; i < 8; ++i)
      rowpart[i] += fmaxf(acc[i], 0.0f) * w2c;
  }

  // reduce across 16 lanes within each half (rows 0-7 / 8-15)
#pragma unroll
  for (int off = 1; off < 16; off <<= 1)
#pragma unroll
    for (int i = 0; i < 8; ++i)
      rowpart[i] += __shfl_xor(rowpart[i], off, 16);

  const float b2v = b2[0];
  if (m == 0) {  // lane 0 -> rows 0-7, lane 16 -> rows 8-15
    const int rb = (lane >= 16) ? 8 : 0;
#pragma unroll
    for (int i = 0; i < 8; ++i)
      lds_mass[wave][rb + i] = softplus_f(rowpart[i] + b2v);
  }
  __syncthreads();

  const int rb = (lane >= 16) ? 8 : 0;
  float mrow[8];
#pragma unroll
  for (int i = 0; i < 8; ++i) mrow[i] = lds_mass[wave][rb + i];

  // ---- phi = c*cos(coords @ rffW + rffb); mass-weighted column sums ----
#pragma unroll
  for (int n = 0; n < 4; ++n) {
    const int col = n * 16 + m;
    const float bias = rffb[col];
    v8f acc;
#pragma unroll
    for (int i = 0; i < 8; ++i) acc[i] = bias;
#pragma unroll
    for (int k = 0; k < 16; ++k) {
      const int kr = 4 * k + koff;
      v2f Bv;
      Bv.x = rffW[kr * ND + col];
      Bv.y = rffW[(kr + 1) * ND + col];
      acc = __builtin_amdgcn_wmma_f32_16x16x4_f32(
          false, A[k], false, Bv, (short)0, acc, false, false);
    }
    float wp = 0.0f;
#pragma unroll
    for (int i = 0; i < 8; ++i)
      wp += RFF_C * cosf(acc[i]) * mrow[i];
    wp += __shfl_xor(wp, 16, 32);  // combine row halves for this column
    if (lane < 16) atomicAdd(&lds_sum[col], wp);
  }
  __syncthreads();

  if (tid < NRFF) {
    const int b = blockIdx.x >> 6;  // 64 blocks (128 tokens each) per batch
    unsafeAtomicAdd(&phi_sum[b * NRFF + tid], lds_sum[tid]);
  }
}

// ---------------- pass 2: grav + scale per token ----------------
__global__ __launch_bounds__(256) void pass2_kernel(
    const float* __restrict__ coords,
    const float* __restrict__ rffW, const float* __restrict__ rffb,
    const float* __restrict__ phi_sum,
    float* __restrict__ scale_arr)
{
  const int tid  = threadIdx.x;
  const int lane = tid & 31;
  const int wave = tid >> 5;
  const int tile = blockIdx.x * WPB + wave;
  const long tok0 = (long)tile * 16;
  const int b    = tile >> 9;  // 512 tiles per batch (8192/16)
  const int m    = lane & 15;
  const int koff = (lane >= 16) ? 2 : 0;

  const float* crow = coords + (tok0 + (long)m) * ND + koff;
  v2f A[16];
#pragma unroll
  for (int k = 0; k < 16; ++k)
    A[k] = *(const v2f*)(crow + 4 * k);

  float gp[8];
#pragma unroll
  for (int i = 0; i < 8; ++i) gp[i] = 0.0f;

#pragma unroll
  for (int n = 0; n < 4; ++n) {
    const int col = n * 16 + m;
    const float bias = rffb[col];
    v8f acc;
#pragma unroll
    for (int i = 0; i < 8; ++i) acc[i] = bias;
#pragma unroll
    for (int k = 0; k < 16; ++k) {
      const int kr = 4 * k + koff;
      v2f Bv;
      Bv.x = rffW[kr * ND + col];
      Bv.y = rffW[(kr + 1) * ND + col];
      acc = __builtin_amdgcn_wmma_f32_16x16x4_f32(
          false, A[k], false, Bv, (short)0, acc, false, false);
    }
    const float S = phi_sum[b * NRFF + col];
#pragma unroll
    for (int i = 0; i < 8; ++i)
      gp[i] += RFF_C * cosf(acc[i]) * S;
  }

#pragma unroll
  for (int off = 1; off < 16; off <<= 1)
#pragma unroll
    for (int i = 0; i < 8; ++i)
      gp[i] += __shfl_xor(gp[i], off, 16);

  if (m == 0) {
    const int rb = (lane >= 16) ? 8 : 0;
#pragma unroll
    for (int i = 0; i < 8; ++i) {
      float g = 0.1f * gp[i];
      g = fminf(fmaxf(g, -0.9f), 5.0f);
      scale_arr[tok0 + rb + i] = sqrtf(1.0f + g + 1e-6f);
    }
  }
}

// ---------------- pass 3: out = U * scale[token], streaming ----------------
__global__ __launch_bounds__(256) void pass3_kernel(
    const v4f* __restrict__ U4,
    const float* __restrict__ scale_arr,
    v4f* __restrict__ out4)
{
  const long nthreads = (long)gridDim.x * blockDim.x;
  long i4 = (long)blockIdx.x * blockDim.x + threadIdx.x;
#pragma unroll
  for (int r = 0; r < 4; ++r) {
    const float s = scale_arr[i4 >> 7];       // 128 float4 per token
    v4f u = __builtin_nontemporal_load(U4 + i4);
    v4f o = u * s;
    __builtin_nontemporal_store(o, out4 + i4);
    i4 += nthreads;
  }
}

extern "C" void kernel_launch(void* const* d_in, const int* in_sizes, int n_in,
                              void* d_out, int out_size, void* d_ws, size_t ws_size,
                              hipStream_t stream) {
  const float* U      = (const float*)d_in[0];
  const float* coords = (const float*)d_in[1];
  const float* w1     = (const float*)d_in[2];
  const float* b1     = (const float*)d_in[3];
  const float* w2     = (const float*)d_in[4];
  const float* b2     = (const float*)d_in[5];
  const float* rffW   = (const float*)d_in[6];
  const float* rffb   = (const float*)d_in[7];
  float* out = (float*)d_out;

  float* phi_sum   = (float*)d_ws;     // 8*64 = 512 floats
  float* scale_arr = phi_sum + 512;    // 8*8192 = 65536 floats

  init_kernel<<<1, 512, 0, stream>>>(phi_sum);
  // B*T/16 = 4096 tiles, 8 waves (tiles) per 256-thread block -> 512 blocks
  pass1_kernel<<<512, 256, 0, stream>>>(coords, w1, b1, w2, b2, rffW, rffb, phi_sum);
  pass2_kernel<<<512, 256, 0, stream>>>(coords, rffW, rffb, phi_sum, scale_arr);
  // 33554432 floats = 8388608 float4; 8192*256 threads * 4 iters
  pass3_kernel<<<8192, 256, 0, stream>>>((const v4f*)U, scale_arr, (v4f*)out);
}